// Axon_layer_84782654423695
// MI455X (gfx1250) — compile-verified
//
#include <hip/hip_runtime.h>
#include <math.h>

// Problem constants (match reference)
#define NS 8
#define NH 256
#define NW 256
#define NB 4
#define NO0 256
#define NO1 256
#define HW (NH * NW)
#define PI_F 3.14159265358979323846f

// to_decard_idx(angle, 256): note (size-1) == 255 for both dims, and the
// reference's "stale size_last" bug is harmless because O0 == O1.
__device__ __forceinline__ float to_decard(float a) {
    float s  = sinf(a);
    float as = asinf(s);
    float b  = cosf(a);
    float c  = sqrtf(1.0f - s * s + 1e-6f);
    float f  = as * b / c;
    f = 2.0f * f / PI_F;
    return (f + 1.0f) * 0.5f * 255.0f;
}

__global__ void zero_out_kernel(float* __restrict__ p, int n) {
    int i = blockIdx.x * blockDim.x + threadIdx.x;
    if (i < n) p[i] = 0.0f;
}

__global__ __launch_bounds__(256) void axon_scatter_kernel(
    const float* __restrict__ signal,   // (B,H,W)
    const float* __restrict__ wsyn,     // (S,H,W)
    const float* __restrict__ coord0,   // (S,H,W)
    const float* __restrict__ coord1,   // (S,H,W)
    float* __restrict__ out)            // (B,O0,O1)
{
    __shared__ float lds_sig[NB][256];

    const int t   = threadIdx.x;
    const int hw0 = blockIdx.x * 256;
    const int hw  = hw0 + t;

    // --- Stage signal[b, hw0:hw0+256] into LDS via CDNA5 async copy ---
    // global_load_async_to_lds_b32: per-lane LDS byte address in VGPR (%0),
    // per-lane 64-bit global address in VGPR pair (%1), no SADDR ("off").
    // Tracked by ASYNCcnt.
#pragma unroll
    for (int b = 0; b < NB; ++b) {
        const float* gptr = signal + b * HW + hw;
        unsigned lds_off = (unsigned)(size_t)(&lds_sig[b][t]); // low 32 bits = LDS offset
        asm volatile("global_load_async_to_lds_b32 %0, %1, off"
                     :: "v"(lds_off), "v"(gptr)
                     : "memory");
    }
#if __has_builtin(__builtin_amdgcn_s_wait_asynccnt)
    __builtin_amdgcn_s_wait_asynccnt(0);
#else
    asm volatile("s_wait_asynccnt 0x0" ::: "memory");
#endif
    __syncthreads();

    // Register-cache the 4 batch signal values for this pixel (reused 72x).
    float sb[NB];
#pragma unroll
    for (int b = 0; b < NB; ++b) sb[b] = lds_sig[b][t];

    const float step = PI_F / 255.0f;  // pi * delta / (size-1), delta in {-1,0,1}

    for (int s = 0; s < NS; ++s) {
        const int base = s * HW + hw;
        // Single-use streams: non-temporal so they don't pollute L2 (output +
        // atomics want to stay resident there).
        const float wv = __builtin_nontemporal_load(wsyn + base);
        const float c0 = __builtin_nontemporal_load(coord0 + base);
        const float c1 = __builtin_nontemporal_load(coord1 + base);

        const float dec0 = to_decard(c0);
        const float dec1 = to_decard(c1);

        int   i0[3], i1[3];
        float r0f[3], r1f[3];
#pragma unroll
        for (int j = 0; j < 3; ++j) {
            float d0 = (j == 1) ? dec0 : to_decard(c0 + (float)(j - 1) * step);
            float d1 = (j == 1) ? dec1 : to_decard(c1 + (float)(j - 1) * step);
            int v0 = __float2int_rn(d0);   // round-half-to-even == jnp.round
            int v1 = __float2int_rn(d1);
            r0f[j] = (float)v0;            // reference uses rounded value in src
            r1f[j] = (float)v1;
            i0[j] = min(max(v0, 0), NO0 - 1);
            i1[j] = min(max(v1, 0), NO1 - 1);
        }

#pragma unroll
        for (int a = 0; a < 3; ++a) {
            const float dx  = r0f[a] - dec0;
            const float dxx = dx * dx;
            const int   row = i0[a] * NO1;
#pragma unroll
            for (int c = 0; c < 3; ++c) {
                const float dy   = r1f[c] - dec1;
                const float srcv = sqrtf(dxx + dy * dy);
                // sigmoid(6*(WW - 2*src)/WW), WW = 1
                const float x    = 6.0f * (1.0f - 2.0f * srcv);
                const float gate = 1.0f / (1.0f + expf(-x));
                const float val  = wv * gate;
                const int   oidx = row + i1[c];
#pragma unroll
                for (int b = 0; b < NB; ++b) {
                    // global_atomic_add_f32, non-returning (STOREcnt path)
                    atomicAdd(out + b * (NO0 * NO1) + oidx, val * sb[b]);
                }
            }
        }
    }
}

extern "C" void kernel_launch(void* const* d_in, const int* in_sizes, int n_in,
                              void* d_out, int out_size, void* d_ws, size_t ws_size,
                              hipStream_t stream) {
    const float* signal = (const float*)d_in[0];   // (B,H,W)
    const float* wsyn   = (const float*)d_in[1];   // (S,H,W)
    const float* coord0 = (const float*)d_in[2];   // (S,H,W)
    const float* coord1 = (const float*)d_in[3];   // (S,H,W)
    float* out = (float*)d_out;                    // (B,O0,O1) = 262144 floats

    // Zero-init output (harness poisons it; scatter-add needs zeros).
    {
        int n = out_size;
        int blocks = (n + 255) / 256;
        zero_out_kernel<<<blocks, 256, 0, stream>>>(out, n);
    }

    // One thread per (h,w) pixel; loops over the 8 synapses internally.
    axon_scatter_kernel<<<HW / 256, 256, 0, stream>>>(signal, wsyn, coord0, coord1, out);
}